// RandomMLPsPenalty_8650064134498
// MI455X (gfx1250) — compile-verified
//
#include <hip/hip_runtime.h>
#include <hip/hip_bf16.h>

// Problem constants (match reference)
#define BDIM 128
#define TDIM 256
#define NDIM 16
#define KDIM 10
#define HDIM 20
#define BT   (BDIM * TDIM)   // 32768 samples per (k,n)

typedef __attribute__((ext_vector_type(2))) float v2f;
typedef __attribute__((ext_vector_type(8))) float v8f;

__device__ __forceinline__ v8f wmma4(v2f a, v2f b, v8f c) {
  // D = A(16x4, f32) * B(4x16, f32) + C(16x16, f32)
  return __builtin_amdgcn_wmma_f32_16x16x4_f32(
      /*neg_a=*/false, a, /*neg_b=*/false, b,
      /*c_mod=*/(short)0, c, /*reuse_a=*/false, /*reuse_b=*/false);
}

// Exchange between the two 16-lane halves of a wave32 (xor-16 pattern).
// v_permlanex16_b32 with identity selects = pure VALU, no DS counter stall.
__device__ __forceinline__ float xhalf16(float v) {
#if __has_builtin(__builtin_amdgcn_permlanex16)
  int i = __builtin_bit_cast(int, v);
  int r = __builtin_amdgcn_permlanex16(i, i, 0x76543210, 0xfedcba98,
                                       /*fi=*/false, /*bound_ctrl=*/false);
  return __builtin_bit_cast(float, r);
#else
  return __shfl_xor(v, 16, 32);
#endif
}

// ---------------------------------------------------------------------------
// K0: per-(t,n) input stats over batch dim (mean, 1/std with ddof=1)
// ---------------------------------------------------------------------------
__global__ void xstats_kernel(const float* __restrict__ inp,
                              float* __restrict__ xmu, float* __restrict__ xrs) {
  int idx = blockIdx.x * blockDim.x + threadIdx.x;  // idx = t*N + n
  if (idx >= TDIM * NDIM) return;
  float s = 0.f, ss = 0.f;
  for (int b = 0; b < BDIM; ++b) {
    float v = inp[b * TDIM * NDIM + idx];
    s += v; ss += v * v;
  }
  float mu  = s * (1.f / BDIM);
  float var = fmaxf((ss - s * mu) * (1.f / (BDIM - 1)), 1e-30f);
  xmu[idx] = mu;
  xrs[idx] = 1.f / sqrtf(var);
}

// ---------------------------------------------------------------------------
// K1: main MLP. One wave per block; block handles one (k,n) pair and 32
// sixteen-row tiles. All elementwise/LN work is done in the WMMA A-fragment
// layout (row across lanes, hidden index across regs + half-wave), so each
// LayerNorm statistic needs only one permlanex16. The 20x20 layer is
// 10 x v_wmma_f32_16x16x4_f32 in full fp32.
// ---------------------------------------------------------------------------
__global__ void __launch_bounds__(32)
mlp_kernel(const float* __restrict__ inp,
           const float* __restrict__ W1, const float* __restrict__ b1,
           const float* __restrict__ W2, const float* __restrict__ b2,
           const float* __restrict__ W3, const float* __restrict__ b3,
           const float* __restrict__ xmu, const float* __restrict__ xrs,
           float* __restrict__ tws) {
  __shared__ float lds[16 * 32];  // D-tile transpose staging (C -> A layout)

  const int lane = threadIdx.x;
  const int mrow = lane & 15;   // sample row within tile (halves mirror)
  const int hi   = lane >> 4;   // half-wave: selects K pair {0,1} vs {2,3}

  const int blk   = blockIdx.x;
  const int chunk = blk & 63;        // 64 chunks of 512 rows
  const int kn    = blk >> 6;
  const int n     = kn & (NDIM - 1);
  const int k     = kn >> 4;
  const int wb    = (k * NDIM + n) * HDIM;

  // Per-lane parameters in A-layout: hidden index i = 4*kc + 2*hi + {0,1}
  float w1x[5], w1y[5], b1x[5], b1y[5];
  float b2x[5], b2y[5], w3x[5], w3y[5];
#pragma unroll
  for (int kc = 0; kc < 5; ++kc) {
    const int i0 = 4 * kc + 2 * hi;  // 0..18, i0+1 <= 19: always valid
    w1x[kc] = W1[wb + i0]; w1y[kc] = W1[wb + i0 + 1];
    b1x[kc] = b1[wb + i0]; b1y[kc] = b1[wb + i0 + 1];
    b2x[kc] = b2[wb + i0]; b2y[kc] = b2[wb + i0 + 1];
    w3x[kc] = W3[wb + i0]; w3y[kc] = W3[wb + i0 + 1];
  }
  const float b3s = b3[k * NDIM + n];

  // B fragments: Bw[i][o] = W2[k][n][o][i], zero-padded for o >= 20.
  // B 4x16 layout: lane holds col o; .x -> K row (4kc+2*hi), .y -> +1.
  v2f Bf[5][2];
  const float* W2p = W2 + (size_t)(k * NDIM + n) * HDIM * HDIM;
#pragma unroll
  for (int kc = 0; kc < 5; ++kc) {
    const int i0 = 4 * kc + 2 * hi;
#pragma unroll
    for (int ct = 0; ct < 2; ++ct) {
      const int o = 16 * ct + mrow;
      float bx = 0.f, by = 0.f;
      if (o < HDIM) { bx = W2p[o * HDIM + i0]; by = W2p[o * HDIM + i0 + 1]; }
      v2f f; f.x = bx; f.y = by;
      Bf[kc][ct] = f;
    }
  }

  float* tout = tws + (size_t)(k * NDIM + n) * BT;

  for (int tix = 0; tix < 32; ++tix) {
    const int rowbase = ((chunk << 5) + tix) << 4;  // 16 rows per tile

    // one standardized x value per lane (its row); halves mirror
    const int rl = rowbase + mrow;
    const int tl = rl & (TDIM - 1);  // row = b*T + t -> t = row % T
    const float xv =
        (inp[rl * NDIM + n] - xmu[tl * NDIM + n]) * xrs[tl * NDIM + n];

    if (tix + 1 < 32) {
      __builtin_prefetch(&inp[(rl + 16) * NDIM + n], 0, 1);
    }

    // ---- layer 1 (ReLU) directly in A-layout ----
    float hx[5], hy[5];
    float s = 0.f;
#pragma unroll
    for (int kc = 0; kc < 5; ++kc) {
      hx[kc] = fmaxf(fmaf(xv, w1x[kc], b1x[kc]), 0.f);
      hy[kc] = fmaxf(fmaf(xv, w1y[kc], b1y[kc]), 0.f);
      s += hx[kc] + hy[kc];
    }
    // ---- LayerNorm over H=20: combine the two half-lanes of each row ----
    s += xhalf16(s);
    const float mu = s * (1.f / HDIM);
    float vs = 0.f;
#pragma unroll
    for (int kc = 0; kc < 5; ++kc) {
      hx[kc] -= mu; hy[kc] -= mu;
      vs += hx[kc] * hx[kc] + hy[kc] * hy[kc];
    }
    vs += xhalf16(vs);
    const float rn = rsqrtf(vs * (1.f / HDIM) + 1e-5f);

    // ---- 16x20 @ 20x20 GEMM: 10 x v_wmma_f32_16x16x4_f32 ----
    v8f acc0 = {0.f, 0.f, 0.f, 0.f, 0.f, 0.f, 0.f, 0.f};
    v8f acc1 = {0.f, 0.f, 0.f, 0.f, 0.f, 0.f, 0.f, 0.f};
#pragma unroll
    for (int kc = 0; kc < 5; ++kc) {
      v2f A; A.x = hx[kc] * rn; A.y = hy[kc] * rn;
      acc0 = wmma4(A, Bf[kc][0], acc0);
      acc1 = wmma4(A, Bf[kc][1], acc1);
    }

    // ---- transpose D tile (C-layout) back to A-layout via LDS ----
#pragma unroll
    for (int v = 0; v < 8; ++v) {
      const int rloc = v + 8 * hi;
      lds[rloc * 32 + mrow] = acc0[v];        // cols 0..15
      lds[rloc * 32 + 16 + mrow] = acc1[v];   // cols 16..31 (20..31 unread)
    }
    __syncthreads();
    float dx[5], dy[5];
    float s2 = 0.f;
#pragma unroll
    for (int kc = 0; kc < 5; ++kc) {
      const v2f d = *(const v2f*)&lds[mrow * 32 + 4 * kc + 2 * hi];
      dx[kc] = fmaxf(d.x + b2x[kc], 0.f);     // bias + ReLU
      dy[kc] = fmaxf(d.y + b2y[kc], 0.f);
      s2 += dx[kc] + dy[kc];
    }
    __syncthreads();

    // ---- LayerNorm 2 + W3 dot, again one permlanex16 per statistic ----
    s2 += xhalf16(s2);
    const float mu2 = s2 * (1.f / HDIM);
    float vs2 = 0.f;
#pragma unroll
    for (int kc = 0; kc < 5; ++kc) {
      dx[kc] -= mu2; dy[kc] -= mu2;
      vs2 += dx[kc] * dx[kc] + dy[kc] * dy[kc];
    }
    vs2 += xhalf16(vs2);
    const float r2 = rsqrtf(vs2 * (1.f / HDIM) + 1e-5f);

    float td = 0.f;
#pragma unroll
    for (int kc = 0; kc < 5; ++kc) {
      td += dx[kc] * w3x[kc] + dy[kc] * w3y[kc];
    }
    td *= r2;
    td += xhalf16(td);  // combine the two half-rows
    const float tv = td + b3s;

    if (lane < 16) tout[rowbase + mrow] = tv;  // coalesced 16-lane store
  }
}

// ---------------------------------------------------------------------------
// K2: per-(k,t,n) stats of t over batch dim (mean, 1/(std+1e-8), ddof=1)
// ---------------------------------------------------------------------------
__global__ void tstats_kernel(const float* __restrict__ tws,
                              float* __restrict__ tmu, float* __restrict__ trs) {
  int idx = blockIdx.x * blockDim.x + threadIdx.x;  // (k*T + t)*N + n
  if (idx >= KDIM * TDIM * NDIM) return;
  int k = idx / (TDIM * NDIM);
  int r = idx - k * TDIM * NDIM;
  int t = r / NDIM;
  int n = r - t * NDIM;
  const float* p = tws + (size_t)(k * NDIM + n) * BT + t;
  float s = 0.f, ss = 0.f;
  for (int b = 0; b < BDIM; ++b) {
    float v = p[b * TDIM];
    s += v; ss += v * v;
  }
  float mu  = s * (1.f / BDIM);
  float var = fmaxf((ss - s * mu) * (1.f / (BDIM - 1)), 0.f);
  tmu[idx] = mu;
  trs[idx] = 1.f / (sqrtf(var) + 1e-8f);
}

// ---------------------------------------------------------------------------
// K3: per-(k,t) marginal / joint products; one thread per (k,t), running
// marginal sums held in registers (no cross-lane ops).
// ---------------------------------------------------------------------------
__global__ void joint_kernel(const float* __restrict__ tws,
                             const float* __restrict__ tmu,
                             const float* __restrict__ trs,
                             float* __restrict__ sq) {
  int kt = blockIdx.x * blockDim.x + threadIdx.x;
  if (kt >= KDIM * TDIM) return;
  int k = kt / TDIM;
  int t = kt - k * TDIM;

  float mu[NDIM], rs[NDIM], ms[NDIM];
#pragma unroll
  for (int n = 0; n < NDIM; ++n) {
    int sidx = (k * TDIM + t) * NDIM + n;
    mu[n] = tmu[sidx];
    rs[n] = trs[sidx];
    ms[n] = 0.f;
  }
  float pj = 0.f;
  for (int b = 0; b < BDIM; ++b) {
    float p = 1.f;
#pragma unroll
    for (int n = 0; n < NDIM; ++n) {
      float val = (tws[(size_t)(k * NDIM + n) * BT + b * TDIM + t] - mu[n]) * rs[n];
      ms[n] += val;
      p *= val;
    }
    pj += p;
  }
  float pm = 1.f;
#pragma unroll
  for (int n = 0; n < NDIM; ++n) pm *= ms[n] * (1.f / BDIM);
  float d = pm - pj * (1.f / BDIM);
  sq[kt] = d * d;
}

// ---------------------------------------------------------------------------
// K4: deterministic final reduction of K*T squared diffs -> scalar mean
// ---------------------------------------------------------------------------
__global__ void final_kernel(const float* __restrict__ sq, float* __restrict__ out) {
  __shared__ float red[256];
  int tid = threadIdx.x;
  float s = 0.f;
  for (int i = tid; i < KDIM * TDIM; i += 256) s += sq[i];
  red[tid] = s;
  __syncthreads();
  for (int off = 128; off > 0; off >>= 1) {
    if (tid < off) red[tid] += red[tid + off];
    __syncthreads();
  }
  if (tid == 0) out[0] = red[0] * (1.f / (KDIM * TDIM));
}

// ---------------------------------------------------------------------------
extern "C" void kernel_launch(void* const* d_in, const int* in_sizes, int n_in,
                              void* d_out, int out_size, void* d_ws, size_t ws_size,
                              hipStream_t stream) {
  (void)in_sizes; (void)n_in; (void)out_size; (void)ws_size;
  const float* inp = (const float*)d_in[0];
  const float* W1  = (const float*)d_in[1];
  const float* b1  = (const float*)d_in[2];
  const float* W2  = (const float*)d_in[3];
  const float* b2  = (const float*)d_in[4];
  const float* W3  = (const float*)d_in[5];
  const float* b3  = (const float*)d_in[6];
  float* out = (float*)d_out;

  // workspace layout (floats): xmu | xrs | t | tmu | trs | sq  (~21.4 MB)
  float* ws  = (float*)d_ws;
  float* xmu = ws;
  float* xrs = xmu + TDIM * NDIM;
  float* tws = xrs + TDIM * NDIM;
  float* tmu = tws + (size_t)KDIM * NDIM * BT;
  float* trs = tmu + (size_t)KDIM * TDIM * NDIM;
  float* sq  = trs + (size_t)KDIM * TDIM * NDIM;

  xstats_kernel<<<(TDIM * NDIM + 255) / 256, 256, 0, stream>>>(inp, xmu, xrs);
  mlp_kernel<<<KDIM * NDIM * (BT / 512), 32, 0, stream>>>(
      inp, W1, b1, W2, b2, W3, b3, xmu, xrs, tws);
  tstats_kernel<<<(KDIM * TDIM * NDIM + 255) / 256, 256, 0, stream>>>(tws, tmu, trs);
  joint_kernel<<<(KDIM * TDIM + 63) / 64, 64, 0, stream>>>(tws, tmu, trs, sq);
  final_kernel<<<1, 256, 0, stream>>>(sq, out);
}